// GraphLearningModule_37194416783904
// MI455X (gfx1250) — compile-verified
//
#include <hip/hip_runtime.h>
#include <math.h>

// Problem constants (match reference)
#define Bm 2
#define Tm 12
#define Nn 4096
#define Kk 16
#define Hh 4
#define Cc 16
#define Vv 3

typedef __attribute__((ext_vector_type(2))) float v2f;
typedef __attribute__((ext_vector_type(8))) float v8f;

// D(16x16,f32) += A(16x4,f32) x B(4x16,f32)
__device__ __forceinline__ v8f wmma4(v2f a, v2f b, v8f c) {
  return __builtin_amdgcn_wmma_f32_16x16x4_f32(false, a, false, b, (short)0, c,
                                               false, false);
}

// x + lane-swizzled(x); imm = (xor_mask<<10) | (or_mask<<5) | and_mask
#define SWZ_ADD(x, imm) \
  ((x) + __int_as_float(__builtin_amdgcn_ds_swizzle(__float_as_int(x), (imm))))
#define SWZ_XOR16 0x401F  // xor 0x10, and 0x1f : swap wave halves
#define SWZ_XOR1  0x041F
#define SWZ_XOR2  0x081F
#define SWZ_XOR4  0x101F
#define SWZ_XOR8  0x201F

// ---------------------------------------------------------------------------
// Kernel 1: one wave per (b,t,n,h). Computes w[k] for the 16 neighbors via
// 4 chained f32 WMMAs (16x16x16), plus deg = sum_k w[k].
// ---------------------------------------------------------------------------
__global__ void spatial_w_deg(const float* __restrict__ feat,
                              const float* __restrict__ multiM,
                              const int* __restrict__ nn,
                              float* __restrict__ u_w,
                              float* __restrict__ deg) {
  const int wid  = blockIdx.x * (blockDim.x >> 5) + (threadIdx.x >> 5);
  const int lane = threadIdx.x & 31;
  const int half = lane >> 4;   // 0: rows 0-7 / K {4s,4s+1}; 1: rows 8-15 / K {4s+2,4s+3}
  const int col  = lane & 15;   // output column = neighbor index k; also A row

  const int h  = wid & (Hh - 1);
  const int n  = (wid >> 2) & (Nn - 1);
  const int bt = wid >> 14;     // b*T + t

  // A fragments: M[h], row = col, element pair (4s+2*half, 4s+2*half+1)
  const float* Mh = multiM + ((size_t)(h * Cc + col)) * Cc + 2 * half;
  v2f a0 = *(const v2f*)(Mh + 0);
  v2f a1 = *(const v2f*)(Mh + 4);
  v2f a2 = *(const v2f*)(Mh + 8);
  v2f a3 = *(const v2f*)(Mh + 12);

  const int  nb    = nn[n * (Kk + 1) + 1 + col];
  const bool valid = (nb >= 0);
  const int  j     = valid ? nb : 0;

  const size_t fbase = (size_t)bt * Nn * Hh * Cc;
  const float* fi = feat + fbase + ((size_t)n * Hh + h) * Cc + 2 * half;
  const float* fj = feat + fbase + ((size_t)j * Hh + h) * Cc + 2 * half;

  v2f b0 = *(const v2f*)(fi + 0)  - *(const v2f*)(fj + 0);
  v2f b1 = *(const v2f*)(fi + 4)  - *(const v2f*)(fj + 4);
  v2f b2 = *(const v2f*)(fi + 8)  - *(const v2f*)(fj + 8);
  v2f b3 = *(const v2f*)(fi + 12) - *(const v2f*)(fj + 12);

  v8f acc = {};
  acc = wmma4(a0, b0, acc);
  acc = wmma4(a1, b1, acc);
  acc = wmma4(a2, b2, acc);
  acc = wmma4(a3, b3, acc);

  // ||Mdf||^2 per column: 8 rows in this lane + 8 rows in lane^16
  float ss = 0.f;
#pragma unroll
  for (int i = 0; i < 8; ++i) ss += acc[i] * acc[i];
  ss = SWZ_ADD(ss, SWZ_XOR16);

  float w = valid ? __expf(-ss) : 0.0f;

  // deg = sum over 16 columns (halves mirror, xor-reduce within 16-groups)
  float dsum = w;
  dsum = SWZ_ADD(dsum, SWZ_XOR1);
  dsum = SWZ_ADD(dsum, SWZ_XOR2);
  dsum = SWZ_ADD(dsum, SWZ_XOR4);
  dsum = SWZ_ADD(dsum, SWZ_XOR8);

  if (lane == 0) deg[wid] = dsum;  // deg index == ((b*T+t)*N+n)*H+h == wid
  if (half == 0) {
    size_t oidx = (((size_t)bt * Nn + n) * Kk + col) * Hh + h;
    u_w[oidx] = w;
  }
}

// ---------------------------------------------------------------------------
// Kernel 2: u_w = w / sqrt(deg_i * deg_j), elementwise with deg_j gather.
// ---------------------------------------------------------------------------
__global__ void spatial_norm(const int* __restrict__ nn,
                             const float* __restrict__ deg,
                             float* __restrict__ u_w) {
  size_t idx = (size_t)blockIdx.x * blockDim.x + threadIdx.x;
  const size_t total = (size_t)Bm * Tm * Nn * Kk * Hh;
  if (idx >= total) return;
  const int h  = (int)(idx) & 3;
  const int k  = (int)(idx >> 2) & 15;
  const int n  = (int)(idx >> 6) & 4095;
  const int bt = (int)(idx >> 18);

  const int nb = nn[n * (Kk + 1) + 1 + k];
  float out = 0.0f;
  if (nb >= 0) {
    const int j = nb & (Nn - 1);  // nbr % N
    float di = deg[((size_t)bt * Nn + n) * Hh + h];
    float dj = deg[((size_t)bt * Nn + j) * Hh + h];
    float dm = sqrtf(di * dj);
    float w  = u_w[idx];
    out = (dm > 0.0f) ? (w / dm) : 0.0f;
  }
  u_w[idx] = out;
}

// ---------------------------------------------------------------------------
// Kernel 3: one wave per (b, tt=0..T-2, h, 16-node block). For each of the 3
// temporal intervals: Qdf via 4 chained WMMAs, wd = sum_i exp(-Qdf_i^2),
// tril mask, normalize by in_deg, store d_w directly.
// ---------------------------------------------------------------------------
__global__ void temporal_dw(const float* __restrict__ feat,
                            const float* __restrict__ multiQ,
                            float* __restrict__ d_w) {
  const int wid  = blockIdx.x * (blockDim.x >> 5) + (threadIdx.x >> 5);
  const int lane = threadIdx.x & 31;
  const int half = lane >> 4;
  const int col  = lane & 15;

  const int nblk = wid & 255;          // N/16 = 256 blocks
  const int h    = (wid >> 8) & 3;
  const int btt  = wid >> 10;
  const int tt   = btt % (Tm - 1);
  const int b    = btt / (Tm - 1);
  const int tj   = tt + 1;
  const int node = nblk * 16 + col;

  const float* fj = feat + ((((size_t)b * Tm + tj) * Nn + node) * Hh + h) * Cc + 2 * half;
  v2f fj0 = *(const v2f*)(fj + 0);
  v2f fj1 = *(const v2f*)(fj + 4);
  v2f fj2 = *(const v2f*)(fj + 8);
  v2f fj3 = *(const v2f*)(fj + 12);

  float wd[Vv];
#pragma unroll
  for (int v = 0; v < Vv; ++v) {
    int ts = tj - (v + 1);
    if (ts < 0) ts += Tm;
    const float* fi = feat + ((((size_t)b * Tm + ts) * Nn + node) * Hh + h) * Cc + 2 * half;
    v2f b0 = *(const v2f*)(fi + 0)  - fj0;
    v2f b1 = *(const v2f*)(fi + 4)  - fj1;
    v2f b2 = *(const v2f*)(fi + 8)  - fj2;
    v2f b3 = *(const v2f*)(fi + 12) - fj3;

    const float* Qh = multiQ + ((size_t)((v * Hh + h) * Cc + col)) * Cc + 2 * half;
    v2f a0 = *(const v2f*)(Qh + 0);
    v2f a1 = *(const v2f*)(Qh + 4);
    v2f a2 = *(const v2f*)(Qh + 8);
    v2f a3 = *(const v2f*)(Qh + 12);

    v8f acc = {};
    acc = wmma4(a0, b0, acc);
    acc = wmma4(a1, b1, acc);
    acc = wmma4(a2, b2, acc);
    acc = wmma4(a3, b3, acc);

    // sum_i exp(-q_i^2)  (exp THEN sum, per reference)
    float es = 0.f;
#pragma unroll
    for (int i = 0; i < 8; ++i) es += __expf(-acc[i] * acc[i]);
    es = SWZ_ADD(es, SWZ_XOR16);

    wd[v] = (v <= tt) ? es : 0.0f;   // tril mask (wave-uniform)
  }

  float ind = wd[0] + wd[1] + wd[2];
  float inv = (ind > 0.0f) ? (1.0f / ind) : 0.0f;

  if (half == 0) {
#pragma unroll
    for (int v = 0; v < Vv; ++v) {
      size_t oidx = ((((size_t)(b * (Tm - 1) + tt) * Vv + v) * Nn + node) * Hh) + h;
      d_w[oidx] = wd[v] * inv;
    }
  }
}

// ---------------------------------------------------------------------------
extern "C" void kernel_launch(void* const* d_in, const int* in_sizes, int n_in,
                              void* d_out, int out_size, void* d_ws, size_t ws_size,
                              hipStream_t stream) {
  const float* feat   = (const float*)d_in[0];  // (B,T,N,H,C) f32
  const float* multiQ = (const float*)d_in[1];  // (V,H,C,C)   f32
  const float* multiM = (const float*)d_in[2];  // (H,C,C)     f32
  const int*   nn     = (const int*)d_in[3];    // (N,K+1)     i32

  float* u_w = (float*)d_out;
  float* dw  = u_w + (size_t)Bm * Tm * Nn * Kk * Hh;
  float* deg = (float*)d_ws;                    // B*T*N*H floats = 1.5 MB

  const int waves1 = Bm * Tm * Nn * Hh;         // 393216 waves, 8/block
  spatial_w_deg<<<waves1 / 8, 256, 0, stream>>>(feat, multiM, nn, u_w, deg);

  const size_t total = (size_t)Bm * Tm * Nn * Kk * Hh;
  spatial_norm<<<(unsigned)((total + 255) / 256), 256, 0, stream>>>(nn, deg, u_w);

  const int waves3 = Bm * (Tm - 1) * Hh * (Nn / 16);  // 22528 waves, 8/block
  temporal_dw<<<waves3 / 8, 256, 0, stream>>>(feat, multiQ, dw);
}